// MultiHeadAttention_20547123544782
// MI455X (gfx1250) — compile-verified
//
#include <hip/hip_runtime.h>

typedef __attribute__((ext_vector_type(2))) float        v2f;
typedef __attribute__((ext_vector_type(8))) float        v8f;
typedef __attribute__((ext_vector_type(4))) unsigned int v4u;
typedef __attribute__((ext_vector_type(4))) int          v4i;
typedef __attribute__((ext_vector_type(8))) int          v8i;

#define D_MODEL 1024
#define SEQ     2048
#define NH      16
#define DH      64
#define SD      (SEQ * D_MODEL)

// LDS layout (dynamic): two buffers, each {A-tile 64x33, B-tile 2112 floats}
#define TILE_F      2112                 // 64*33 (also covers 32*64 packed B)
#define BUF_F       (2 * TILE_F)         // 4224 floats per buffer
#define LDS_BYTES   (2 * BUF_F * 4)      // 33792 B, double buffered

#if __has_builtin(__builtin_amdgcn_tensor_load_to_lds)
#define HAVE_TDM 1
#else
#define HAVE_TDM 0
#endif

#if HAVE_TDM
// ---------------------------------------------------------------------------
// Issue one TDM 2D tile load: R rows x W elems (fp32), row stride `strideElems`,
// into LDS byte offset ldsOff. Optional hardware row padding (pad_enable with
// pad_interval=2^(code+1) dwords, pad_amount=1 dword) gives [R][W+1] layout.
// Descriptor packing per CDNA5 ISA ch.8 (D# groups 0..3).
// ---------------------------------------------------------------------------
__device__ __forceinline__ void tdm_load_tile(const float* gptr,
                                              unsigned ldsOff, int W, int R,
                                              long strideElems,
                                              int padEnable, int padCode)
{
    unsigned long long addr = (unsigned long long)gptr;
    v4u g0;
    g0.x = 1u;                                        // count=1, user mode
    g0.y = ldsOff;                                    // lds_addr (bytes)
    g0.z = (unsigned)(addr & 0xFFFFFFFFu);            // global_addr[31:0]
    g0.w = (unsigned)((addr >> 32) & 0x01FFFFFFu)     // global_addr[56:32]
         | 0x80000000u;                               // type=2 (image)
    v8i g1;
    g1[0] = (int)((2u << 16)                          // data_size = 4B
          | (padEnable ? ((1u << 20) | ((unsigned)padCode << 22)) : 0u));
    g1[1] = (int)((unsigned)W << 16);                 // tensor_dim0[15:0]
    g1[2] = (int)((unsigned)R << 16);                 // tensor_dim1[15:0]
    g1[3] = (int)((unsigned)W << 16);                 // tile_dim0
    g1[4] = (int)(unsigned)R;                         // tile_dim1 (tile_dim2=0)
    g1[5] = (int)(unsigned)(strideElems & 0xFFFFFFFFu);       // dim0_stride lo
    g1[6] = (int)((unsigned)((strideElems >> 32) & 0xFFFFu)); // dim0_stride hi
    g1[7] = 0;
    v4i z4 = {0, 0, 0, 0};
#if defined(__clang_major__) && (__clang_major__ >= 23)
    v8i z8 = {0, 0, 0, 0, 0, 0, 0, 0};
    __builtin_amdgcn_tensor_load_to_lds(g0, g1, z4, z4, z8, 0);
#else
    __builtin_amdgcn_tensor_load_to_lds(g0, g1, z4, z4, 0);
#endif
}

__device__ __forceinline__ void wait_tensorcnt0()
{
#if __has_builtin(__builtin_amdgcn_s_wait_tensorcnt)
    __builtin_amdgcn_s_wait_tensorcnt(0);
#else
    asm volatile("s_wait_tensorcnt 0x0" ::: "memory");
#endif
}
#endif // HAVE_TDM

// ---------------------------------------------------------------------------
// 64x64-output-tile fp32 GEMM using V_WMMA_F32_16X16X4_F32.
// A-tile staged as [64][33]; B-tile as [32][64] (normal) or [64][33] (btrans,
// read transposed at the WMMA operand build -> no LDS-store transpose needed).
// TDM double-buffered pipeline: one workgroup barrier per 32-wide K step.
// ---------------------------------------------------------------------------
__device__ __forceinline__ void gemm64x64_f32(
    const float* __restrict__ A, int lda,
    const float* __restrict__ B, int ldb, bool btrans,
    float* __restrict__ C, int ldc,
    const float* __restrict__ bias, float scale,
    int Kdim, int m0, int n0)
{
    extern __shared__ float lds[];

    const int tid  = threadIdx.x;
    const int lane = tid & 31;
    const int w    = tid >> 5;     // wave 0..7
    const int wm   = w >> 2;       // 0..1
    const int wn   = w & 3;        // 0..3
    const int half = lane >> 4;    // 0..1
    const int r    = lane & 15;

    const int nIter = Kdim >> 5;
    v8f acc0 = {};
    v8f acc1 = {};

#if HAVE_TDM
    auto issue = [&](int it, int buf) {
        const unsigned base = (unsigned)buf * (BUF_F * 4u);
        const int k0 = it << 5;
        // A tile: 64 rows x 32, HW-padded to stride 33 (padCode 4 = 32 dwords)
        tdm_load_tile(A + (long)m0 * lda + k0, base, 32, 64, lda, 1, 4);
        if (!btrans)   // B tile: 32 rows x 64, packed
            tdm_load_tile(B + (long)k0 * ldb + n0, base + TILE_F * 4u,
                          64, 32, ldb, 0, 0);
        else           // B^T tile: 64 rows x 32, padded to 33 (same shape as A)
            tdm_load_tile(B + (long)n0 * ldb + k0, base + TILE_F * 4u,
                          32, 64, ldb, 1, 4);
    };

    if (w == 0) issue(0, 0);
#endif

    for (int it = 0; it < nIter; ++it) {
        const int buf = it & 1;
        const float* As = lds + buf * BUF_F;
        const float* Bs = As + TILE_F;

#if HAVE_TDM
        if (w == 0) wait_tensorcnt0();          // current buffer landed
        __syncthreads();                        // publish + retire prev reads
        if (w == 0 && it + 1 < nIter) issue(it + 1, buf ^ 1);
#else
        const int k0 = it << 5;
        float* Asw = lds + buf * BUF_F;
        float* Bsw = Asw + TILE_F;
        #pragma unroll
        for (int i = 0; i < 8; ++i) {
            int e = tid + i * 256;
            int row = e >> 5, col = e & 31;
            Asw[row * 33 + col] = A[(long)(m0 + row) * lda + (k0 + col)];
        }
        if (!btrans) {
            #pragma unroll
            for (int i = 0; i < 8; ++i) {
                int e = tid + i * 256;
                int row = e >> 6, col = e & 63;
                Bsw[row * 64 + col] = B[(long)(k0 + row) * ldb + (n0 + col)];
            }
        } else {
            #pragma unroll
            for (int i = 0; i < 8; ++i) {
                int e = tid + i * 256;
                int n = e >> 5, k = e & 31;
                Bsw[n * 33 + k] = B[(long)(n0 + n) * ldb + (k0 + k)];
            }
        }
        __syncthreads();
#endif

        #pragma unroll
        for (int kk = 0; kk < 32; kk += 4) {
            const int ka = kk + 2 * half;       // lanes16-31 carry k+2 (ISA layout)
            v2f a0, a1, b;
            a0.x = As[(wm * 32 +      r) * 33 + ka];
            a0.y = As[(wm * 32 +      r) * 33 + ka + 1];
            a1.x = As[(wm * 32 + 16 + r) * 33 + ka];
            a1.y = As[(wm * 32 + 16 + r) * 33 + ka + 1];
            if (!btrans) {
                b.x = Bs[ ka      * 64 + wn * 16 + r];
                b.y = Bs[(ka + 1) * 64 + wn * 16 + r];
            } else {
                b.x = Bs[(wn * 16 + r) * 33 + ka];
                b.y = Bs[(wn * 16 + r) * 33 + ka + 1];
            }
            acc0 = __builtin_amdgcn_wmma_f32_16x16x4_f32(
                       false, a0, false, b, (short)0, acc0, false, false);
            acc1 = __builtin_amdgcn_wmma_f32_16x16x4_f32(
                       false, a1, false, b, (short)0, acc1, false, false);
        }
#if !HAVE_TDM
        __syncthreads();
#endif
    }

    const int col = n0 + wn * 16 + r;
    const float bb = bias ? bias[col] : 0.0f;
    #pragma unroll
    for (int i = 0; i < 8; ++i) {
        int row0 = m0 + wm * 32 +      i + 8 * half;
        int row1 = m0 + wm * 32 + 16 + i + 8 * half;
        C[(long)row0 * ldc + col] = acc0[i] * scale + bb;
        C[(long)row1 * ldc + col] = acc1[i] * scale + bb;
    }
}

// ---------------------------------------------------------------------------
// Stage kernels
// ---------------------------------------------------------------------------
__global__ __launch_bounds__(256) void k_proj(
    const float* __restrict__ X, const float* __restrict__ W,
    const float* __restrict__ bias, float* __restrict__ Y)
{
    gemm64x64_f32(X, D_MODEL, W, D_MODEL, false, Y, D_MODEL,
                  bias, 1.0f, D_MODEL, blockIdx.y * 64, blockIdx.x * 64);
}

__global__ __launch_bounds__(256) void k_scores(
    const float* __restrict__ Q, const float* __restrict__ Kp,
    float* __restrict__ attn)
{
    const int z = blockIdx.z;
    const int b = z >> 4, h = z & 15;
    gemm64x64_f32(Q  + (long)b * SD + h * DH, D_MODEL,
                  Kp + (long)b * SD + h * DH, D_MODEL, true,
                  attn + (long)z * SEQ * SEQ, SEQ,
                  nullptr, 0.125f, DH, blockIdx.y * 64, blockIdx.x * 64);
}

__global__ __launch_bounds__(256) void k_softmax(float* __restrict__ attn)
{
    __shared__ float red[256];
    float* p = attn + (long)blockIdx.x * SEQ;
    const int t = threadIdx.x;

    float v[8];
    float mx = -3.402823466e38f;
    #pragma unroll
    for (int i = 0; i < 8; ++i) {
        v[i] = p[t + i * 256];
        mx = fmaxf(mx, v[i]);
    }
    red[t] = mx; __syncthreads();
    for (int s = 128; s > 0; s >>= 1) {
        if (t < s) red[t] = fmaxf(red[t], red[t + s]);
        __syncthreads();
    }
    mx = red[0];
    __syncthreads();

    float sum = 0.0f;
    #pragma unroll
    for (int i = 0; i < 8; ++i) {
        v[i] = expf(v[i] - mx);
        sum += v[i];
    }
    red[t] = sum; __syncthreads();
    for (int s = 128; s > 0; s >>= 1) {
        if (t < s) red[t] += red[t + s];
        __syncthreads();
    }
    const float inv = 1.0f / red[0];
    #pragma unroll
    for (int i = 0; i < 8; ++i)
        p[t + i * 256] = v[i] * inv;
}

__global__ __launch_bounds__(256) void k_context(
    const float* __restrict__ attn, const float* __restrict__ V,
    float* __restrict__ ctx)
{
    const int z = blockIdx.z;
    const int b = z >> 4, h = z & 15;
    gemm64x64_f32(attn + (long)z * SEQ * SEQ, SEQ,
                  V + (long)b * SD + h * DH, D_MODEL, false,
                  ctx + (long)b * SD + h * DH, D_MODEL,
                  nullptr, 1.0f, SEQ, blockIdx.y * 64, blockIdx.x * 64);
}

// ---------------------------------------------------------------------------
// Host launch
// ---------------------------------------------------------------------------
extern "C" void kernel_launch(void* const* d_in, const int* in_sizes, int n_in,
                              void* d_out, int out_size, void* d_ws, size_t ws_size,
                              hipStream_t stream)
{
    const float* query = (const float*)d_in[0];
    const float* key   = (const float*)d_in[1];
    const float* value = (const float*)d_in[2];
    const float* Wq    = (const float*)d_in[3];
    const float* bq    = (const float*)d_in[4];
    const float* Wk    = (const float*)d_in[5];
    const float* bk    = (const float*)d_in[6];
    const float* Wv    = (const float*)d_in[7];
    const float* bv    = (const float*)d_in[8];
    const float* Wo    = (const float*)d_in[9];
    const float* bo    = (const float*)d_in[10];

    const long actElems = 2L * SD;
    float* Q   = (float*)d_ws;
    float* Kp  = Q  + actElems;
    float* V   = Kp + actElems;
    float* ctx = Q;                          // Q dead after k_scores -> reuse
    float* out  = (float*)d_out;
    float* attn = out + actElems;

    const dim3 blk(256);
    const dim3 gProj(D_MODEL / 64, (2 * SEQ) / 64);
    const dim3 gScore(SEQ / 64, SEQ / 64, 2 * NH);
    const dim3 gCtx(DH / 64, SEQ / 64, 2 * NH);

    k_proj<<<gProj, blk, LDS_BYTES, stream>>>(query, Wq, bq, Q);
    k_proj<<<gProj, blk, LDS_BYTES, stream>>>(key,   Wk, bk, Kp);
    k_proj<<<gProj, blk, LDS_BYTES, stream>>>(value, Wv, bv, V);
    k_scores<<<gScore, blk, LDS_BYTES, stream>>>(Q, Kp, attn);
    k_softmax<<<2 * NH * SEQ, blk, 0, stream>>>(attn);
    k_context<<<gCtx, blk, LDS_BYTES, stream>>>(attn, V, ctx);
    k_proj<<<gProj, blk, LDS_BYTES, stream>>>(ctx, Wo, bo, out);
}